// TamiyoPolicyGNN_11991548691170
// MI455X (gfx1250) — compile-verified
//
#include <hip/hip_runtime.h>
#include <math.h>

typedef float v2f __attribute__((ext_vector_type(2)));
typedef float v8f __attribute__((ext_vector_type(8)));

#define NEG_INF (-3.402823466e+38f)

constexpr int Nn = 100000;   // nodes
constexpr int Ee = 1600000;  // edges
constexpr int Ff = 64;       // input feature dim
constexpr int Hh = 128;      // hidden dim
constexpr int Gg = 64;       // graphs
constexpr int Ll = 3;        // gcn layers

// ---------------------------------------------------------------- utilities

__global__ void k_fill(float* __restrict__ p, float v, int n) {
    int i = blockIdx.x * blockDim.x + threadIdx.x;
    if (i < n) p[i] = v;
}

__global__ void k_deg(const int* __restrict__ dst, float* __restrict__ deg, int E) {
    int i = blockIdx.x * blockDim.x + threadIdx.x;
    if (i < E) atomicAdd(&deg[dst[i]], 1.0f);
}

__global__ void k_rsqrt(float* __restrict__ deg, int n) {
    int i = blockIdx.x * blockDim.x + threadIdx.x;
    if (i < n) deg[i] = rsqrtf(deg[i]);
}

__device__ inline void atomicMaxF(float* addr, float val) {
    unsigned int* ua = (unsigned int*)addr;
    unsigned int old = __float_as_uint(*addr);
    while (__uint_as_float(old) < val) {
        unsigned int assumed = old;
        old = atomicCAS(ua, assumed, __float_as_uint(val));
        if (old == assumed) break;
    }
}

// ------------------------------------------------- WMMA fp32 GEMM (Ncol=128)
// C[M,128] = act(A[M,K] @ B[K,128] + bias)
// One 16x16 C tile per wave32 via V_WMMA_F32_16X16X4_F32.
// ISA 7.12.2 layouts:
//   A 16x4 : lanes 0-15 -> M=lane, {v0:K=k, v1:K=k+1};  lanes 16-31 -> {K=k+2,K=k+3}
//   B 4x16 : lanes 0-15 -> N=lane, {v0:K=k, v1:K=k+1};  lanes 16-31 -> {K=k+2,K=k+3}
//   C/D    : vgpr r -> lanes 0-15 (M=r, N=lane), lanes 16-31 (M=r+8, N=lane-16)
template <int K>
__global__ void k_gemm_wmma(const float* __restrict__ A, const float* __restrict__ B,
                            const float* __restrict__ bias, float* __restrict__ C,
                            int M, int do_relu) {
    constexpr int NCOL = 128;
    constexpr int SBS = 132;      // padded LDS stride (bank-conflict avoidance)
    constexpr int SAS = K + 4;
    __shared__ float sB[K * SBS];
    __shared__ float sA[16 * SAS];

    // stage the whole weight matrix B (<= 64 KB) into LDS once per block
    for (int i = threadIdx.x; i < K * NCOL; i += blockDim.x) {
        int r = i >> 7, c = i & 127;
        sB[r * SBS + c] = B[i];
    }

    const int wave = threadIdx.x >> 5;
    const int lane = threadIdx.x & 31;
    const int hi   = lane >> 4;     // 0: K=k,k+1  1: K=k+2,k+3
    const int l16  = lane & 15;
    const int col0 = wave << 4;     // 8 waves cover all 128 output cols

    const int nRowTiles = M >> 4;
    for (int rt = blockIdx.x; rt < nRowTiles; rt += gridDim.x) {
        const int row0 = rt << 4;
        __syncthreads();
        // stage 16 x K strip of A, coalesced along K
        for (int i = threadIdx.x; i < 16 * K; i += blockDim.x) {
            int r = i / K, c = i % K;
            sA[r * SAS + c] = A[(long)(row0 + r) * K + c];
        }
        __syncthreads();

        v8f acc = {};
#pragma unroll
        for (int k = 0; k < K; k += 4) {
            const int ka = k + 2 * hi;
            v2f a, b;
            a.x = sA[l16 * SAS + ka];
            a.y = sA[l16 * SAS + ka + 1];
            b.x = sB[ka * SBS + col0 + l16];
            b.y = sB[(ka + 1) * SBS + col0 + l16];
            acc = __builtin_amdgcn_wmma_f32_16x16x4_f32(
                false, a, false, b, (short)0, acc, false, false);
        }

        const int ccol  = col0 + l16;
        const int rbase = row0 + hi * 8;
        const float bv  = bias ? bias[ccol] : 0.0f;
#pragma unroll
        for (int r = 0; r < 8; ++r) {
            float v = acc[r] + bv;
            if (do_relu) v = fmaxf(v, 0.0f);
            C[(long)(rbase + r) * NCOL + ccol] = v;
        }
    }
}

// ------------------------------------------------------------- GCN kernels

// agg[i,h] = gcn_b[h] + xw[i,h] * dinv[i]^2   (bias + self-loop term)
__global__ void k_agg_init(const float* __restrict__ xw, const float* __restrict__ dinv,
                           const float* __restrict__ bias, float* __restrict__ agg,
                           int total) {
    int i = blockIdx.x * blockDim.x + threadIdx.x;
    if (i >= total) return;
    int node = i >> 7, h = i & 127;
    float di = dinv[node];
    agg[i] = bias[h] + xw[i] * di * di;
}

// one wave32 per edge: lane handles 4 columns (float4 = full 128B/row per 8 lanes)
__global__ void k_scatter(const int* __restrict__ src, const int* __restrict__ dst,
                          const float* __restrict__ dinv, const float* __restrict__ xw,
                          float* __restrict__ agg, int E) {
    long e = (long)blockIdx.x * (blockDim.x >> 5) + (threadIdx.x >> 5);
    if (e >= E) return;
    int lane = threadIdx.x & 31;
    int s = src[e], d = dst[e];
    float nrm = dinv[s] * dinv[d];
    float4 v = ((const float4*)(xw + (long)s * Hh))[lane];
    float* ap = agg + (long)d * Hh + lane * 4;
    atomicAdd(ap + 0, v.x * nrm);
    atomicAdd(ap + 1, v.y * nrm);
    atomicAdd(ap + 2, v.z * nrm);
    atomicAdd(ap + 3, v.w * nrm);
}

__global__ void k_residual_relu(float* __restrict__ x, const float* __restrict__ agg,
                                int total) {
    int i = blockIdx.x * blockDim.x + threadIdx.x;
    if (i < total) x[i] += fmaxf(agg[i], 0.0f);
}

// ------------------------------------------------------------- pooling

__global__ void k_pool_init(float* __restrict__ gsum, float* __restrict__ gmax,
                            float* __restrict__ cnts) {
    int i = blockIdx.x * blockDim.x + threadIdx.x;
    if (i < Gg * Hh) { gsum[i] = 0.0f; gmax[i] = NEG_INF; }
    if (i < Gg) cnts[i] = 0.0f;
}

__global__ void k_pool(const float* __restrict__ x, const int* __restrict__ batch,
                       float* __restrict__ gsum, float* __restrict__ gmax,
                       float* __restrict__ cnts, int N) {
    long n = (long)blockIdx.x * (blockDim.x >> 5) + (threadIdx.x >> 5);
    if (n >= N) return;
    int lane = threadIdx.x & 31;
    int b = batch[n];
    float4 v = ((const float4*)(x + n * Hh))[lane];
    float* sp = gsum + (long)b * Hh + lane * 4;
    atomicAdd(sp + 0, v.x); atomicAdd(sp + 1, v.y);
    atomicAdd(sp + 2, v.z); atomicAdd(sp + 3, v.w);
    float* mp = gmax + (long)b * Hh + lane * 4;
    atomicMaxF(mp + 0, v.x); atomicMaxF(mp + 1, v.y);
    atomicMaxF(mp + 2, v.z); atomicMaxF(mp + 3, v.w);
    if (lane == 0) atomicAdd(&cnts[b], 1.0f);
}

__global__ void k_repr(const float* __restrict__ gsum, const float* __restrict__ gmax,
                       const float* __restrict__ cnts, float* __restrict__ repr) {
    int i = blockIdx.x * blockDim.x + threadIdx.x;
    if (i >= Gg * Hh) return;
    int g = i >> 7, h = i & 127;
    float c = fmaxf(cnts[g], 1.0f);
    repr[g * 256 + h]       = gsum[i] / c;
    repr[g * 256 + 128 + h] = gmax[i];
}

// ------------------------------------------------------------- heads (tiny)

__global__ void k_head(const float* __restrict__ repr,
                       const float* __restrict__ dw1, const float* __restrict__ db1,
                       const float* __restrict__ dw2, const float* __restrict__ db2,
                       const float* __restrict__ dw3, const float* __restrict__ db3,
                       const float* __restrict__ vw1, const float* __restrict__ vb1,
                       const float* __restrict__ vw2, const float* __restrict__ vb2,
                       float* __restrict__ out) {
    __shared__ float sr[256];
    __shared__ float h1[128];
    __shared__ float h2[64];
    __shared__ float v1[128];
    __shared__ float red[128];
    const int g = blockIdx.x;
    const int t = threadIdx.x;  // 128 threads

    sr[t] = repr[g * 256 + t];
    sr[t + 128] = repr[g * 256 + 128 + t];
    __syncthreads();

    float acc = db1[t];
    float accv = vb1[t];
    for (int k = 0; k < 256; ++k) {
        float r = sr[k];
        acc  += r * dw1[k * 128 + t];
        accv += r * vw1[k * 128 + t];
    }
    h1[t] = fmaxf(acc, 0.0f);
    v1[t] = fmaxf(accv, 0.0f);
    __syncthreads();

    if (t < 64) {
        float a2 = db2[t];
        for (int k = 0; k < 128; ++k) a2 += h1[k] * dw2[k * 64 + t];
        h2[t] = fmaxf(a2, 0.0f);
    }
    __syncthreads();

    if (t < 3) {
        float a3 = db3[t];
        for (int k = 0; k < 64; ++k) a3 += h2[k] * dw3[k * 3 + t];
        if (t == 0) out[g]          = 1.0f / (1.0f + expf(-a3));   // adaptation_prob
        if (t == 1) out[Gg + g]     = 1.0f;                        // softmax of 1 logit
        if (t == 2) out[2 * Gg + g] = 1.0f / (1.0f + expf(-a3));   // urgency
    }

    red[t] = v1[t] * vw2[t];
    __syncthreads();
    for (int s = 64; s > 0; s >>= 1) {
        if (t < s) red[t] += red[t + s];
        __syncthreads();
    }
    if (t == 0) out[3 * Gg + g] = red[0] + vb2[0];                 // value
}

// ------------------------------------------------------------- launcher

extern "C" void kernel_launch(void* const* d_in, const int* in_sizes, int n_in,
                              void* d_out, int out_size, void* d_ws, size_t ws_size,
                              hipStream_t stream) {
    const float* node_features = (const float*)d_in[0];
    const int*   edge_index    = (const int*)d_in[1];
    const int*   batch         = (const int*)d_in[2];
    const float* enc_w1 = (const float*)d_in[3];
    const float* enc_b1 = (const float*)d_in[4];
    const float* enc_w2 = (const float*)d_in[5];
    const float* enc_b2 = (const float*)d_in[6];
    const float* gcn_w  = (const float*)d_in[7];
    const float* gcn_b  = (const float*)d_in[8];
    const float* dec_w1 = (const float*)d_in[9];
    const float* dec_b1 = (const float*)d_in[10];
    const float* dec_w2 = (const float*)d_in[11];
    const float* dec_b2 = (const float*)d_in[12];
    const float* dec_w3 = (const float*)d_in[13];
    const float* dec_b3 = (const float*)d_in[14];
    const float* val_w1 = (const float*)d_in[15];
    const float* val_b1 = (const float*)d_in[16];
    const float* val_w2 = (const float*)d_in[17];
    const float* val_b2 = (const float*)d_in[18];
    float* out = (float*)d_out;

    const int* e_src = edge_index;
    const int* e_dst = edge_index + Ee;

    // workspace layout (floats; all offsets multiple of 4 for float4 access)
    float* ws   = (float*)d_ws;
    float* dinv = ws;                          // N   (deg -> dinv in place)
    float* xbuf = dinv + Nn;                   // N*H
    float* t0   = xbuf + (long)Nn * Hh;        // N*H (x1 / xw)
    float* agg  = t0 + (long)Nn * Hh;          // N*H
    float* gsum = agg + (long)Nn * Hh;         // G*H
    float* gmax = gsum + Gg * Hh;              // G*H
    float* cnts = gmax + Gg * Hh;              // G
    float* repr = cnts + Gg;                   // G*2H

    const int NH = Nn * Hh;

    // degree with self-loops -> dinv
    k_fill<<<(Nn + 255) / 256, 256, 0, stream>>>(dinv, 1.0f, Nn);
    k_deg<<<(Ee + 255) / 256, 256, 0, stream>>>(e_dst, dinv, Ee);
    k_rsqrt<<<(Nn + 255) / 256, 256, 0, stream>>>(dinv, Nn);

    // encoder
    k_gemm_wmma<64><<<1024, 256, 0, stream>>>(node_features, enc_w1, enc_b1, t0, Nn, 1);
    k_gemm_wmma<128><<<1024, 256, 0, stream>>>(t0, enc_w2, enc_b2, xbuf, Nn, 0);

    // GCN layers
    for (int l = 0; l < Ll; ++l) {
        k_gemm_wmma<128><<<1024, 256, 0, stream>>>(xbuf, gcn_w + (long)l * Hh * Hh,
                                                   nullptr, t0, Nn, 0);
        k_agg_init<<<(NH + 255) / 256, 256, 0, stream>>>(t0, dinv, gcn_b + l * Hh, agg, NH);
        k_scatter<<<(Ee + 7) / 8, 256, 0, stream>>>(e_src, e_dst, dinv, t0, agg, Ee);
        k_residual_relu<<<(NH + 255) / 256, 256, 0, stream>>>(xbuf, agg, NH);
    }

    // pooling
    k_pool_init<<<(Gg * Hh + 255) / 256, 256, 0, stream>>>(gsum, gmax, cnts);
    k_pool<<<(Nn + 7) / 8, 256, 0, stream>>>(xbuf, batch, gsum, gmax, cnts, Nn);
    k_repr<<<(Gg * Hh + 255) / 256, 256, 0, stream>>>(gsum, gmax, cnts, repr);

    // heads -> outputs [adaptation_prob | layer_priorities | urgency | value]
    k_head<<<Gg, 128, 0, stream>>>(repr, dec_w1, dec_b1, dec_w2, dec_b2, dec_w3, dec_b3,
                                   val_w1, val_b1, val_w2, val_b2, out);
    (void)in_sizes; (void)n_in; (void)out_size; (void)ws_size;
}